// PyramidROIAlign_23656679867535
// MI455X (gfx1250) — compile-verified
//
#include <hip/hip_runtime.h>
#include <cstdint>
#include <cstddef>

// ---------------------------------------------------------------------------
// Pyramid ROI Align for MI455X (gfx1250).
// Bandwidth-bound gather kernel: 16 taps (b128 each) per pooled output pixel,
// staged to LDS via gfx1250 async global->LDS DMA (ASYNCcnt) when available.
// ---------------------------------------------------------------------------

#if defined(__gfx1250__) && __has_builtin(__builtin_amdgcn_global_load_async_to_lds_b128)
#define USE_ASYNC_LDS 1
#else
#define USE_ASYNC_LDS 0
#endif

#if USE_ASYNC_LDS
typedef int v4i __attribute__((vector_size(16)));
typedef __attribute__((address_space(1))) v4i* gv4i_p;   // global int4*
typedef __attribute__((address_space(3))) v4i* lv4i_p;   // LDS int4*
#endif

__device__ __forceinline__ float4 ld_f4(const float* p) {
    return *reinterpret_cast<const float4*>(p);
}

extern "C" __global__ __launch_bounds__(256, 1) void pyramid_roi_align_gfx1250(
    const float* __restrict__ rois,   // [B*N, 4]  y1 x1 y2 x2
    const float* __restrict__ f2,     // [B, H2,   H2,   256]
    const float* __restrict__ f3,     // [B, H2/2, H2/2, 256]
    const float* __restrict__ f4,
    const float* __restrict__ f5,
    const float* __restrict__ meta,   // [B, 11]; [6]=H, [7]=W
    float* __restrict__ out,          // [B*N, 7, 7, 256]
    int nroi, int N, int H2)
{
    extern __shared__ float4 stage[];            // 16 taps * 256 threads = 64 KB
    const int tid = threadIdx.x;
    const int gid = blockIdx.x * 256 + tid;
    const int pix = gid >> 6;                    // pooled pixel id (roi*49 + py*7 + px)
    const int c4  = (gid & 63) << 2;             // channel base (float4)
    if (pix >= nroi * 49) return;

    const int roi = pix / 49;
    const int pp  = pix - roi * 49;
    const int py  = pp / 7;
    const int px  = pp - py * 7;
    const int b   = roi / N;

    const float y1 = rois[roi * 4 + 0];
    const float x1 = rois[roi * 4 + 1];
    const float y2 = rois[roi * 4 + 2];
    const float x2 = rois[roi * 4 + 3];
    const float Him = meta[6];
    const float Wim = meta[7];

    // --- FPN level selection (one-hot in the reference) ---
    const float hh   = fmaxf(0.0f, y2 - y1);
    const float ww   = fmaxf(0.0f, x2 - x1);
    const float area = sqrtf(ww * hh + 1e-8f);
    float lvl = floorf(4.0f + log2f(area * (1.0f / 224.0f)));
    lvl = fminf(fmaxf(lvl, 2.0f), 5.0f);
    const int li = (int)lvl - 2;
    const int Hf = H2 >> li;                     // square feature maps
    const float* __restrict__ fp = (li == 0) ? f2 : (li == 1) ? f3 : (li == 2) ? f4 : f5;
    const float scale = (float)(Hf - 1);

    const float y1n = y1 / Him, y2n = y2 / Him;
    const float x1n = x1 / Wim, x2n = x2 / Wim;

    // --- sampling coordinates for the 2x2 crop window feeding this pooled px ---
    int   ry[4], cx[4];
    float fy[2], fx[2], vy[2], vx[2];
#pragma unroll
    for (int k = 0; k < 2; ++k) {
        const float ty = (float)(2 * py + k) * (1.0f / 13.0f);
        const float ys = (y1n + (y2n - y1n) * ty) * scale;
        const float yf = floorf(ys);
        fy[k] = ys - yf;
        ry[2 * k + 0] = (int)fminf(fmaxf(yf,        0.0f), scale);
        ry[2 * k + 1] = (int)fminf(fmaxf(yf + 1.0f, 0.0f), scale);
        vy[k] = (ys >= 0.0f && ys <= scale) ? 1.0f : 0.0f;

        const float tx = (float)(2 * px + k) * (1.0f / 13.0f);
        const float xs = (x1n + (x2n - x1n) * tx) * scale;
        const float xf = floorf(xs);
        fx[k] = xs - xf;
        cx[2 * k + 0] = (int)fminf(fmaxf(xf,        0.0f), scale);
        cx[2 * k + 1] = (int)fminf(fmaxf(xf + 1.0f, 0.0f), scale);
        vx[k] = (xs >= 0.0f && xs <= scale) ? 1.0f : 0.0f;
    }

    // --- gather the 16 taps (4 rows x 4 cols), 4 channels each ---
    float4 tap[16];
#if USE_ASYNC_LDS
#pragma unroll
    for (int r = 0; r < 4; ++r) {
#pragma unroll
        for (int c = 0; c < 4; ++c) {
            const int k = r * 4 + c;
            const float* src = fp + (((size_t)(b * Hf + ry[r]) * Hf + cx[c]) << 8) + c4;
            __builtin_amdgcn_global_load_async_to_lds_b128(
                (gv4i_p)src, (lv4i_p)&stage[k * 256 + tid], /*offset=*/0, /*cpol=*/0);
        }
    }
#if __has_builtin(__builtin_amdgcn_s_wait_asynccnt)
    __builtin_amdgcn_s_wait_asynccnt(0);
#else
    asm volatile("s_wait_asynccnt 0" ::: "memory");
#endif
#pragma unroll
    for (int k = 0; k < 16; ++k) tap[k] = stage[k * 256 + tid];
#else
#pragma unroll
    for (int r = 0; r < 4; ++r)
#pragma unroll
        for (int c = 0; c < 4; ++c)
            tap[r * 4 + c] = ld_f4(fp + (((size_t)(b * Hf + ry[r]) * Hf + cx[c]) << 8) + c4);
#endif

    // --- bilinear blend of the 2x2 crop pixels, then max-pool them ---
    float4 v[4];
#pragma unroll
    for (int iy = 0; iy < 2; ++iy) {
#pragma unroll
        for (int jx = 0; jx < 2; ++jx) {
            const float m   = vy[iy] * vx[jx];
            const float w00 = (1.0f - fy[iy]) * (1.0f - fx[jx]) * m;
            const float w01 = (1.0f - fy[iy]) * fx[jx] * m;
            const float w10 = fy[iy] * (1.0f - fx[jx]) * m;
            const float w11 = fy[iy] * fx[jx] * m;
            const float4 t00 = tap[(2 * iy)     * 4 + 2 * jx];
            const float4 t01 = tap[(2 * iy)     * 4 + 2 * jx + 1];
            const float4 t10 = tap[(2 * iy + 1) * 4 + 2 * jx];
            const float4 t11 = tap[(2 * iy + 1) * 4 + 2 * jx + 1];
            float4 r;
            r.x = t00.x * w00 + t01.x * w01 + t10.x * w10 + t11.x * w11;
            r.y = t00.y * w00 + t01.y * w01 + t10.y * w10 + t11.y * w11;
            r.z = t00.z * w00 + t01.z * w01 + t10.z * w10 + t11.z * w11;
            r.w = t00.w * w00 + t01.w * w01 + t10.w * w10 + t11.w * w11;
            v[iy * 2 + jx] = r;
        }
    }
    float4 best;
    best.x = fmaxf(fmaxf(v[0].x, v[1].x), fmaxf(v[2].x, v[3].x));
    best.y = fmaxf(fmaxf(v[0].y, v[1].y), fmaxf(v[2].y, v[3].y));
    best.z = fmaxf(fmaxf(v[0].z, v[1].z), fmaxf(v[2].z, v[3].z));
    best.w = fmaxf(fmaxf(v[0].w, v[1].w), fmaxf(v[2].w, v[3].w));

    *reinterpret_cast<float4*>(out + (((size_t)pix) << 8) + c4) = best;
}

extern "C" void kernel_launch(void* const* d_in, const int* in_sizes, int n_in,
                              void* d_out, int out_size, void* d_ws, size_t ws_size,
                              hipStream_t stream) {
    const float* rois = (const float*)d_in[0];
    const float* f2   = (const float*)d_in[1];
    const float* f3   = (const float*)d_in[2];
    const float* f4   = (const float*)d_in[3];
    const float* f5   = (const float*)d_in[4];
    const float* meta = (const float*)d_in[5];
    float* out = (float*)d_out;

    const int nroi = in_sizes[0] / 4;          // B*N
    const int B    = in_sizes[5] / 11;
    const int N    = nroi / B;
    // feat2 is [B, H2, H2, 256]: solve H2^2 = size/(B*256)
    const long long hw = (long long)in_sizes[1] / ((long long)B * 256);
    int H2 = 1;
    while ((long long)(H2 + 1) * (H2 + 1) <= hw) ++H2;

    const long long total_threads = (long long)nroi * 49 * 64;
    const int blocks = (int)((total_threads + 255) / 256);
    const size_t lds_bytes = 16 * 256 * sizeof(float4);   // 64 KB stage

    pyramid_roi_align_gfx1250<<<blocks, 256, lds_bytes, stream>>>(
        rois, f2, f3, f4, f5, meta, out, nroi, N, H2);
}